// HeadTailAggregation_5059471475159
// MI455X (gfx1250) — compile-verified
//
#include <hip/hip_runtime.h>
#include <cstdint>
#include <cstddef>

// ---------------------------------------------------------------------------
// HeadTailAggregation for MI455X (gfx1250, wave32)
//
// out[g, 0:64]   = x[head(g), :]   where head(g) = first i with index[i]==g
// out[g, 64:128] = x[tail(g), :]   where tail(g) = last  i with index[i]==g
//
// index is a sorted PyG-style batch vector covering all G groups, so
// head/tail are segment boundaries -> detect with one neighbor compare.
// Total traffic ~110 MB -> ~5 us at 23.3 TB/s; purely bandwidth bound.
// ---------------------------------------------------------------------------

// Kernel 1: boundary detection. Every group appears at least once, so every
// head[]/tail[] slot is written (no init needed; d_ws arrives poisoned).
__global__ void HeadTail_find_bounds(const int* __restrict__ index,
                                     int* __restrict__ head,
                                     int* __restrict__ tail,
                                     int N) {
  int i = blockIdx.x * blockDim.x + threadIdx.x;
  if (i >= N) return;
  int g = index[i];
  if (i == 0) head[g] = 0;
  if (i == N - 1) {
    tail[g] = N - 1;
  } else {
    int gn = index[i + 1];
    if (gn != g) {
      tail[g] = i;       // last row of segment g
      head[gn] = i + 1;  // first row of segment gn
    }
  }
}

// Kernel 2 (D==64 fast path): one wave32 per group.
// Lanes 0..15 move the 256B head row, lanes 16..31 the 256B tail row,
// 16B per lane via the gfx1250 async copy engine (global -> LDS, ASYNCcnt),
// then a coalesced 512B-per-wave global_store_b128 burst into out.
__global__ void __launch_bounds__(256)
HeadTail_gather_async(const float* __restrict__ x,
                      const int* __restrict__ head,
                      const int* __restrict__ tail,
                      float* __restrict__ out,
                      int G) {
  __shared__ __align__(16) float lds[8 * 128];  // 512 B per wave, 4 KB/block

  const int t    = blockIdx.x * blockDim.x + threadIdx.x;
  const int g    = t >> 5;               // one group per wave32
  const int lane = threadIdx.x & 31;
  const int wv   = threadIdx.x >> 5;
  if (g >= G) return;                    // full blocks when G%8==0 (launcher guarantees)

  const int h  = head[g];
  const int tl = tail[g];

  // Each lane owns one 16-byte chunk: head row for lanes 0..15, tail row after.
  const float* src = (lane < 16) ? (x + (size_t)h  * 64 + (size_t)lane * 4)
                                 : (x + (size_t)tl * 64 + (size_t)(lane - 16) * 4);

  float* lslot = &lds[wv * 128 + lane * 4];
  // addrspace(3) -> generic keeps the LDS byte offset in the low 32 bits.
  uint32_t lds_off = (uint32_t)(uintptr_t)lslot;

  // gfx1250 async copy: memory -> LDS, no VGPR staging, tracked by ASYNCcnt.
  asm volatile("global_load_async_to_lds_b128 %0, %1, off"
               :
               : "v"(lds_off), "v"(src)
               : "memory");
  asm volatile("s_wait_asynccnt 0" ::: "memory");

  // Read back own 16B chunk and store: out viewed as [G][32] float4,
  // writes are perfectly coalesced (512B per wave).
  float4 v = *reinterpret_cast<const float4*>(lslot);
  reinterpret_cast<float4*>(out)[(size_t)g * 32 + lane] = v;
}

// Generic fallback for arbitrary D (scalar, still fully coalesced on out).
__global__ void HeadTail_gather_direct(const float* __restrict__ x,
                                       const int* __restrict__ head,
                                       const int* __restrict__ tail,
                                       float* __restrict__ out,
                                       long total, int D) {
  long t = blockIdx.x * (long)blockDim.x + threadIdx.x;
  if (t >= total) return;
  int twoD = 2 * D;
  long row = t / twoD;
  int  c   = (int)(t - row * twoD);
  int  src = (c < D) ? head[row] : tail[row];
  int  cc  = (c < D) ? c : c - D;
  out[t] = x[(size_t)src * D + cc];
}

extern "C" void kernel_launch(void* const* d_in, const int* in_sizes, int n_in,
                              void* d_out, int out_size, void* d_ws, size_t ws_size,
                              hipStream_t stream) {
  const float* x     = (const float*)d_in[0];   // [N, D] f32
  const int*   index = (const int*)d_in[1];     // [N] i32 (sorted group ids)
  // d_in[2] is the scalar `size` on device; G is derived on host instead
  // (no device reads allowed under graph capture).

  const int N = in_sizes[1];
  const int D = in_sizes[0] / N;                // 64
  const int G = out_size / (2 * D);             // 100000

  int* head = (int*)d_ws;                       // [G]
  int* tail = head + G;                         // [G]  (2*G*4 = 800 KB scratch)
  float* out = (float*)d_out;

  // Pass 1: segment boundaries.
  HeadTail_find_bounds<<<(N + 255) / 256, 256, 0, stream>>>(index, head, tail, N);

  // Pass 2: gather + concat.
  if (D == 64 && (G % 8) == 0) {
    HeadTail_gather_async<<<G / 8, 256, 0, stream>>>(x, head, tail, out, G);
  } else {
    long total = (long)G * 2 * D;
    HeadTail_gather_direct<<<(int)((total + 255) / 256), 256, 0, stream>>>(
        x, head, tail, out, total, D);
  }
}